// PurifiedGraphEncoder_721554505999
// MI455X (gfx1250) — compile-verified
//
#include <hip/hip_runtime.h>
#include <hip/hip_bf16.h>

#define Nn 50000
#define Ee 640000
#define Hh 128
#define Cc 40

typedef __attribute__((ext_vector_type(2))) float v2f;
typedef __attribute__((ext_vector_type(8))) float v8f;

// ---------------------------------------------------------------------------
// WMMA fp32 GEMM: out[N x 128] = act( (HAS_SCALE? diag(rs):I) * A @ W + bias )
// Block = 256 threads = 8 wave32; block covers 16 rows x 128 cols.
// Wave w computes the 16x16 tile at columns [16w, 16w+16).
// A tile staged in LDS (pad->conflict-free column reads); W staged in LDS in
// pair-interleaved layout so each lane's B fragment {W[ka][n],W[ka+1][n]} is a
// single ds_load_b64. Two K-phases keep LDS at ~45 KB. K loop fully unrolled
// so every LDS load uses a constant immediate offset (no induction math).
// ---------------------------------------------------------------------------
#define BSTRIDE 144  // pair-row stride: 2*144 mod 64 == 32 -> halves hit disjoint banks

template <bool RELU, bool HAS_BIAS, bool ACCUM, bool HAS_SCALE>
__global__ __launch_bounds__(256) void gemm128_wmma(
    const float* __restrict__ A, const float* __restrict__ W,
    const float* __restrict__ bias, const float* __restrict__ rowscale,
    float* __restrict__ out)
{
    __shared__ float  As[16][132];
    __shared__ float2 Bs[32 * BSTRIDE];  // 32 pair-rows (=64 K values) per phase

    const int tid  = threadIdx.x;
    const int row0 = blockIdx.x * 16;

    const int wave = tid >> 5;
    const int lane = tid & 31;
    const int col0 = wave * 16;
    const int half = lane >> 4;   // 0: K {k,k+1}, 1: K {k+2,k+3}
    const int l16  = lane & 15;

    // ---- stage A tile (16x128), optionally row-scaled -----------------------
    {
        const int r  = tid >> 4;          // 0..15
        const int c8 = (tid & 15) * 8;    // 0..120
        float s = 1.0f;
        if (HAS_SCALE) s = rowscale[row0 + r];
        const float* p = A + (size_t)(row0 + r) * Hh + c8;
        float4 u0 = *(const float4*)(p);
        float4 u1 = *(const float4*)(p + 4);
        As[r][c8 + 0] = u0.x * s; As[r][c8 + 1] = u0.y * s;
        As[r][c8 + 2] = u0.z * s; As[r][c8 + 3] = u0.w * s;
        As[r][c8 + 4] = u1.x * s; As[r][c8 + 5] = u1.y * s;
        As[r][c8 + 6] = u1.z * s; As[r][c8 + 7] = u1.w * s;
    }

    // ---- init accumulator ---------------------------------------------------
    v8f c = {0.f, 0.f, 0.f, 0.f, 0.f, 0.f, 0.f, 0.f};
    if (ACCUM) {
        // C/D layout: VGPR j holds (m = j + 8*half, n = l16)
        #pragma unroll
        for (int j = 0; j < 8; ++j)
            c[j] = out[(size_t)(row0 + j + 8 * half) * Hh + col0 + l16];
    }
    float bv = 0.0f;
    if (HAS_BIAS) bv = bias[col0 + l16];   // one load; column-invariant over j

    // A fragment base: row M=l16, starting at K offset 2*half
    const float* ap = &As[l16][2 * half];
    // B fragment base: pair-row `half`, column col0+l16
    const float2* bp2 = &Bs[half * BSTRIDE + col0 + l16];

    // ---- main loop: two phases of K=64 --------------------------------------
    #pragma unroll 1
    for (int phase = 0; phase < 2; ++phase) {
        if (phase) __syncthreads();  // protect Bs reuse against previous reads
        {   // stage pair-interleaved W: Bs[p][n] = {W[2p'][n], W[2p'+1][n]}
            const int p  = tid >> 3;          // 0..31 (pair-row within phase)
            const int c0 = (tid & 7) * 4;     // 0..28
            const float* w0 = W + (size_t)(64 * phase + 2 * p) * Hh;
            const float* w1 = w0 + Hh;
            #pragma unroll
            for (int i = 0; i < 4; ++i) {
                const int n = c0 + 32 * i;
                float4 a4 = *(const float4*)(w0 + n);
                float4 b4 = *(const float4*)(w1 + n);
                float2* d = &Bs[p * BSTRIDE + n];
                d[0] = make_float2(a4.x, b4.x);
                d[1] = make_float2(a4.y, b4.y);
                d[2] = make_float2(a4.z, b4.z);
                d[3] = make_float2(a4.w, b4.w);
            }
        }
        __syncthreads();  // also covers As on phase 0

        const float* apk = ap + 64 * phase;

        // fully unrolled: all DS addresses are base + constant immediate offset
        #pragma unroll
        for (int k = 0; k < 64; k += 4) {
            v2f a, b;
            // A 16x4 fragment: lane -> row M=l16, regs hold K=ka, ka+1
            a[0] = apk[k];
            a[1] = apk[k + 1];
            // B 4x16 fragment: lane -> col N=l16, regs hold K=ka, ka+1
            float2 bp = bp2[(k >> 1) * BSTRIDE];
            b[0] = bp.x;
            b[1] = bp.y;
            c = __builtin_amdgcn_wmma_f32_16x16x4_f32(
                /*neg_a=*/false, a, /*neg_b=*/false, b,
                /*c_mod=*/(short)0, c, /*reuse_a=*/false, /*reuse_b=*/false);
        }
    }

    // ---- epilogue (branch-free) ---------------------------------------------
    #pragma unroll
    for (int j = 0; j < 8; ++j) {
        float v = c[j];
        if (HAS_BIAS) v += bv;
        if (RELU) v = fmaxf(v, 0.0f);
        out[(size_t)(row0 + j + 8 * half) * Hh + col0 + l16] = v;
    }
}

// ---------------------------------------------------------------------------
// LayerNorm over H=128: one wave per row; optional fused (resid + relu(x)).
// ---------------------------------------------------------------------------
template <bool HAS_RESID>
__global__ __launch_bounds__(256) void layernorm_kernel(
    const float* __restrict__ x, const float* __restrict__ resid,
    const float* __restrict__ g, const float* __restrict__ bvec,
    float* __restrict__ out)
{
    const int wave = threadIdx.x >> 5;
    const int lane = threadIdx.x & 31;
    const int row  = blockIdx.x * 8 + wave;
    if (row >= Nn) return;

    float4 v = *(const float4*)(x + (size_t)row * Hh + lane * 4);
    if (HAS_RESID) {
        float4 r = *(const float4*)(resid + (size_t)row * Hh + lane * 4);
        v.x = r.x + fmaxf(v.x, 0.f);
        v.y = r.y + fmaxf(v.y, 0.f);
        v.z = r.z + fmaxf(v.z, 0.f);
        v.w = r.w + fmaxf(v.w, 0.f);
    }

    float s = v.x + v.y + v.z + v.w;
    #pragma unroll
    for (int off = 16; off >= 1; off >>= 1) s += __shfl_xor(s, off, 32);
    const float mu = s * (1.0f / 128.0f);

    const float dx = v.x - mu, dy = v.y - mu, dz = v.z - mu, dw = v.w - mu;
    float q = dx * dx + dy * dy + dz * dz + dw * dw;
    #pragma unroll
    for (int off = 16; off >= 1; off >>= 1) q += __shfl_xor(q, off, 32);
    const float rstd = rsqrtf(q * (1.0f / 128.0f) + 1e-5f);

    float4 gg = *(const float4*)(g + lane * 4);
    float4 bb = *(const float4*)(bvec + lane * 4);
    float4 o;
    o.x = dx * rstd * gg.x + bb.x;
    o.y = dy * rstd * gg.y + bb.y;
    o.z = dz * rstd * gg.z + bb.z;
    o.w = dw * rstd * gg.w + bb.w;
    *(float4*)(out + (size_t)row * Hh + lane * 4) = o;
}

// ---------------------------------------------------------------------------
// Graph plumbing
// ---------------------------------------------------------------------------
__global__ void zero_kernel(float4* __restrict__ p, int n4) {
    int i = blockIdx.x * blockDim.x + threadIdx.x;
    if (i < n4) p[i] = make_float4(0.f, 0.f, 0.f, 0.f);
}

__global__ void degree_kernel(const long long* __restrict__ dst, float* __restrict__ cnt) {
    int e = blockIdx.x * blockDim.x + threadIdx.x;
    if (e < Ee) atomicAdd(&cnt[(int)dst[e]], 1.0f);
}

__global__ void invcnt_kernel(float* __restrict__ cnt) {
    int i = blockIdx.x * blockDim.x + threadIdx.x;
    if (i < Nn) {
        float c = cnt[i];
        cnt[i] = (c > 0.f) ? (1.0f / c) : 0.0f;
    }
}

// one wave per edge: 32 lanes x float4 covers the 128-wide message
__global__ __launch_bounds__(256) void scatter_add_kernel(
    const float* __restrict__ h, const long long* __restrict__ src,
    const long long* __restrict__ dst, float* __restrict__ msg)
{
    int gid = blockIdx.x * blockDim.x + threadIdx.x;
    int e = gid >> 5;
    if (e >= Ee) return;
    int seg = (gid & 31) * 4;
    int s = (int)src[e];
    int d = (int)dst[e];
    float4 v = *(const float4*)(h + (size_t)s * Hh + seg);
    float* mp = msg + (size_t)d * Hh + seg;
    atomicAdd(mp + 0, v.x);
    atomicAdd(mp + 1, v.y);
    atomicAdd(mp + 2, v.z);
    atomicAdd(mp + 3, v.w);
}

// logits[N,40] = t[N,128] @ W[128,40] + b ; one wave per row
__global__ __launch_bounds__(256) void cls_out_kernel(
    const float* __restrict__ t, const float* __restrict__ W,
    const float* __restrict__ b, float* __restrict__ logits)
{
    const int wave = threadIdx.x >> 5;
    const int lane = threadIdx.x & 31;
    const int row  = blockIdx.x * 8 + wave;
    if (row >= Nn) return;
    const float* tp = t + (size_t)row * Hh;
    const int c1 = 32 + (lane & 7);  // second column for lanes 0..7
    float acc0 = 0.f, acc1 = 0.f;
    #pragma unroll 4
    for (int k = 0; k < Hh; ++k) {
        float tk = tp[k];
        acc0 = fmaf(tk, W[k * Cc + lane], acc0);
        acc1 = fmaf(tk, W[k * Cc + c1], acc1);
    }
    logits[(size_t)row * Cc + lane] = acc0 + b[lane];
    if (lane < 8) logits[(size_t)row * Cc + c1] = acc1 + b[c1];
}

// ---------------------------------------------------------------------------
extern "C" void kernel_launch(void* const* d_in, const int* in_sizes, int n_in,
                              void* d_out, int out_size, void* d_ws, size_t ws_size,
                              hipStream_t stream)
{
    const float*     x       = (const float*)d_in[0];
    const long long* ei      = (const long long*)d_in[1];
    const float*     proj_w1 = (const float*)d_in[2];
    const float*     proj_b1 = (const float*)d_in[3];
    const float*     proj_w2 = (const float*)d_in[4];
    const float*     proj_b2 = (const float*)d_in[5];
    const float*     in_g    = (const float*)d_in[6];
    const float*     in_b    = (const float*)d_in[7];
    const float*     sage_wl = (const float*)d_in[8];   // [L,H,H]
    const float*     sage_bl = (const float*)d_in[9];   // [L,H]
    const float*     sage_wr = (const float*)d_in[10];  // [L,H,H]
    const float*     ln_g    = (const float*)d_in[11];  // [L,H]
    const float*     ln_b    = (const float*)d_in[12];  // [L,H]
    const float*     cls_w1  = (const float*)d_in[13];
    const float*     cls_b1  = (const float*)d_in[14];
    const float*     cls_w2  = (const float*)d_in[15];  // [H,C]
    const float*     cls_b2  = (const float*)d_in[16];

    const long long* src = ei;        // edge_index[0]
    const long long* dst = ei + Ee;   // edge_index[1]

    // workspace carve-out
    char* ws = (char*)d_ws;
    float* inv_cnt = (float*)ws;              ws += ((size_t)Nn * 4 + 255) & ~(size_t)255;
    const size_t NH = (size_t)Nn * Hh;
    float* b0 = (float*)ws;                   ws += NH * 4;
    float* b1 = (float*)ws;                   ws += NH * 4;
    float* b2 = (float*)ws;                   ws += NH * 4;

    float* logits = (float*)d_out;
    float* h_out  = (float*)d_out + (size_t)Nn * Cc;

    const int gemm_blocks = Nn / 16;                 // 3125 (exact)
    const int ln_blocks   = (Nn + 7) / 8;            // wave-per-row
    const int nh4         = (int)(NH / 4);           // 1.6M float4

    // 1) degree -> inv_cnt
    zero_kernel<<<(Nn / 4 + 255) / 256, 256, 0, stream>>>((float4*)inv_cnt, Nn / 4);
    degree_kernel<<<(Ee + 255) / 256, 256, 0, stream>>>(dst, inv_cnt);
    invcnt_kernel<<<(Nn + 255) / 256, 256, 0, stream>>>(inv_cnt);

    // 2) input projection MLP + LayerNorm
    gemm128_wmma<true, true, false, false><<<gemm_blocks, 256, 0, stream>>>(
        x, proj_w1, proj_b1, nullptr, b0);
    gemm128_wmma<false, true, false, false><<<gemm_blocks, 256, 0, stream>>>(
        b0, proj_w2, proj_b2, nullptr, b1);
    layernorm_kernel<false><<<ln_blocks, 256, 0, stream>>>(b1, nullptr, in_g, in_b, b0);

    // 3) SAGE layers (ping-pong: h lives in b0, then b1; b2 = h_next)
    // layer 0: h = b0
    zero_kernel<<<(nh4 + 255) / 256, 256, 0, stream>>>((float4*)b1, nh4);
    scatter_add_kernel<<<(Ee * 32) / 256, 256, 0, stream>>>(b0, src, dst, b1);
    gemm128_wmma<false, true, false, true><<<gemm_blocks, 256, 0, stream>>>(
        b1, sage_wl, sage_bl, inv_cnt, b2);
    gemm128_wmma<false, false, true, false><<<gemm_blocks, 256, 0, stream>>>(
        b0, sage_wr, nullptr, nullptr, b2);
    layernorm_kernel<true><<<ln_blocks, 256, 0, stream>>>(b2, b0, ln_g, ln_b, b1);

    // layer 1: h = b1, write normalized result straight into d_out's h region
    zero_kernel<<<(nh4 + 255) / 256, 256, 0, stream>>>((float4*)b0, nh4);
    scatter_add_kernel<<<(Ee * 32) / 256, 256, 0, stream>>>(b1, src, dst, b0);
    gemm128_wmma<false, true, false, true><<<gemm_blocks, 256, 0, stream>>>(
        b0, sage_wl + (size_t)Hh * Hh, sage_bl + Hh, inv_cnt, b2);
    gemm128_wmma<false, false, true, false><<<gemm_blocks, 256, 0, stream>>>(
        b1, sage_wr + (size_t)Hh * Hh, nullptr, nullptr, b2);
    layernorm_kernel<true><<<ln_blocks, 256, 0, stream>>>(b2, b1, ln_g + Hh, ln_b + Hh, h_out);

    // 4) classifier
    gemm128_wmma<true, true, false, false><<<gemm_blocks, 256, 0, stream>>>(
        h_out, cls_w1, cls_b1, nullptr, b0);
    cls_out_kernel<<<ln_blocks, 256, 0, stream>>>(b0, cls_w2, cls_b2, logits);
}